// EdgeConv_33930241638504
// MI455X (gfx1250) — compile-verified
//
#include <hip/hip_runtime.h>

typedef __attribute__((ext_vector_type(2))) float v2f;
typedef __attribute__((ext_vector_type(8))) float v8f;

#define IN_F   16
#define OUT_F  16
#define T_F    8
#define KTOT   144          // 8*16 (eattr x h outer product) + 16 (h for edge-bias)
#define KSTEPS 36           // 144 / 4
#define TILE   16
#define EDGE_WAVES 8
#define LN_EPS 1e-5f

// ---------------------------------------------------------------------------
// Kernel 1: pack Wbig[144][16]:
//   rows 0..127 : Wbig[t*16+i][o] = w_edge[t][i*16+o]
//   rows 128..143: Wbig[128+i][o] = b_edge[i*16+o]
// ---------------------------------------------------------------------------
__global__ void build_wbig(const float* __restrict__ w_edge,
                           const float* __restrict__ b_edge,
                           float* __restrict__ wbig) {
    for (int idx = threadIdx.x; idx < KTOT * OUT_F; idx += blockDim.x) {
        int r = idx >> 4, o = idx & 15;
        float v;
        if (r < 128) {
            int t = r >> 4, i = r & 15;
            v = w_edge[t * 256 + i * 16 + o];
        } else {
            v = b_edge[(r - 128) * 16 + o];
        }
        wbig[idx] = v;
    }
}

// ---------------------------------------------------------------------------
// Kernel 2: per-node LayerNorm + ReLU -> h ; out = h @ root + bias
// ---------------------------------------------------------------------------
__global__ void __launch_bounds__(256)
node_pre(const float* __restrict__ x, const float* __restrict__ gamma,
         const float* __restrict__ beta, const float* __restrict__ root,
         const float* __restrict__ bias, float* __restrict__ h,
         float* __restrict__ out, int N) {
    int n = blockIdx.x * blockDim.x + threadIdx.x;
    if (n >= N) return;

    float xv[16];
    const float4* xp = (const float4*)(x + (size_t)n * 16);
    float4 q0 = xp[0], q1 = xp[1], q2 = xp[2], q3 = xp[3];
    xv[0]=q0.x; xv[1]=q0.y; xv[2]=q0.z; xv[3]=q0.w;
    xv[4]=q1.x; xv[5]=q1.y; xv[6]=q1.z; xv[7]=q1.w;
    xv[8]=q2.x; xv[9]=q2.y; xv[10]=q2.z; xv[11]=q2.w;
    xv[12]=q3.x; xv[13]=q3.y; xv[14]=q3.z; xv[15]=q3.w;

    float mu = 0.f;
#pragma unroll
    for (int i = 0; i < 16; ++i) mu += xv[i];
    mu *= (1.f / 16.f);
    float var = 0.f;
#pragma unroll
    for (int i = 0; i < 16; ++i) { float d = xv[i] - mu; var += d * d; }
    var *= (1.f / 16.f);
    float rs = rsqrtf(var + LN_EPS);

    float hv[16];
#pragma unroll
    for (int i = 0; i < 16; ++i) {
        float t = (xv[i] - mu) * rs * gamma[i] + beta[i];
        hv[i] = t > 0.f ? t : 0.f;
    }

    float4* hp = (float4*)(h + (size_t)n * 16);
    hp[0] = make_float4(hv[0], hv[1], hv[2], hv[3]);
    hp[1] = make_float4(hv[4], hv[5], hv[6], hv[7]);
    hp[2] = make_float4(hv[8], hv[9], hv[10], hv[11]);
    hp[3] = make_float4(hv[12], hv[13], hv[14], hv[15]);

    float ov[16];
#pragma unroll
    for (int o = 0; o < 16; ++o) ov[o] = bias[o];
#pragma unroll
    for (int i = 0; i < 16; ++i) {
        float hi = hv[i];
#pragma unroll
        for (int o = 0; o < 16; ++o) ov[o] = fmaf(hi, root[i * 16 + o], ov[o]);
    }

    float4* op = (float4*)(out + (size_t)n * 16);
    op[0] = make_float4(ov[0], ov[1], ov[2], ov[3]);
    op[1] = make_float4(ov[4], ov[5], ov[6], ov[7]);
    op[2] = make_float4(ov[8], ov[9], ov[10], ov[11]);
    op[3] = make_float4(ov[12], ov[13], ov[14], ov[15]);
}

// ---------------------------------------------------------------------------
// Kernel 3: fused edge GEMM + scatter.
// One wave = one 16-edge tile. msg(16x16) = Z(16x144) @ Wbig(144x16) via
// 36 x v_wmma_f32_16x16x4_f32, then global_atomic_add_f32 scatter.
// ---------------------------------------------------------------------------
#define SH_PITCH 20   // padded row strides (bank-conflict-free, 16B aligned)
#define SE_PITCH 12

__global__ void __launch_bounds__(256)
edge_conv(const int* __restrict__ eidx,      // [2, E]
          const float* __restrict__ eattr,   // [E, 8]
          const float* __restrict__ h,       // [N, 16]
          const float* __restrict__ wbig,    // [144, 16]
          float* __restrict__ out,           // [N, 16]
          int E) {
    __shared__ float s_wb[KTOT * OUT_F];
    __shared__ float s_h[EDGE_WAVES][TILE][SH_PITCH];
    __shared__ float s_e[EDGE_WAVES][TILE][SE_PITCH];
    __shared__ int   s_dst[EDGE_WAVES][TILE];

    const int tid  = threadIdx.x;
    const int wave = tid >> 5;
    const int lane = tid & 31;
    const int l15  = lane & 15;

    // stage Wbig into LDS (whole block)
    for (int i = tid; i < KTOT * OUT_F; i += 256) s_wb[i] = wbig[i];

    const int tile0 = (blockIdx.x * EDGE_WAVES + wave) * TILE;

    if (lane < TILE) {
        int e = tile0 + lane;
        if (e < E) {
            int src = eidx[e];
            int dst = eidx[E + e];
            const float4* hp = (const float4*)(h + (size_t)src * 16);
            float4 h0 = hp[0], h1 = hp[1], h2 = hp[2], h3 = hp[3];
            float4* shp = (float4*)&s_h[wave][lane][0];
            shp[0] = h0; shp[1] = h1; shp[2] = h2; shp[3] = h3;
            const float4* ep = (const float4*)(eattr + (size_t)e * 8);
            float4 e0 = ep[0], e1 = ep[1];
            float4* sep = (float4*)&s_e[wave][lane][0];
            sep[0] = e0; sep[1] = e1;
            s_dst[wave][lane] = dst;
        } else {
            float4 z4 = make_float4(0.f, 0.f, 0.f, 0.f);
            float4* shp = (float4*)&s_h[wave][lane][0];
            shp[0] = z4; shp[1] = z4; shp[2] = z4; shp[3] = z4;
            float4* sep = (float4*)&s_e[wave][lane][0];
            sep[0] = z4; sep[1] = z4;
            s_dst[wave][lane] = -1;
        }
    }
    __syncthreads();

    // Fragment layout (f32, K=4): lanes 0-15 hold K = {4k,4k+1},
    // lanes 16-31 hold K = {4k+2,4k+3}; A row = lane&15, B col = lane&15.
    const int half  = lane >> 4;       // 0 or 1
    const int khalf = half * 2;

    v8f acc = {0.f, 0.f, 0.f, 0.f, 0.f, 0.f, 0.f, 0.f};

#pragma unroll
    for (int k = 0; k < KSTEPS; ++k) {
        const int j0 = 4 * k + khalf;
        const int j1 = j0 + 1;
        float a0, a1;
        if (k < 32) {  // outer-product part: z[m][j] = eattr[m][j>>4] * h[m][j&15]
            a0 = s_e[wave][l15][j0 >> 4] * s_h[wave][l15][j0 & 15];
            a1 = s_e[wave][l15][j1 >> 4] * s_h[wave][l15][j1 & 15];
        } else {       // augmented part: plain h (picks up edge-bias rows of Wbig)
            a0 = s_h[wave][l15][j0 - 128];
            a1 = s_h[wave][l15][j1 - 128];
        }
        v2f A; A[0] = a0; A[1] = a1;
        v2f B; B[0] = s_wb[j0 * 16 + l15]; B[1] = s_wb[j1 * 16 + l15];
        acc = __builtin_amdgcn_wmma_f32_16x16x4_f32(
            /*neg_a=*/false, A, /*neg_b=*/false, B,
            /*c_mod=*/(short)0, acc, /*reuse_a=*/false, /*reuse_b=*/false);
    }

    // C/D layout: VGPR r -> tile row r + 8*half, column = lane&15
    const int rbase = half * 8;
#pragma unroll
    for (int r = 0; r < 8; ++r) {
        int dst = s_dst[wave][rbase + r];
        if (dst >= 0) atomicAdd(&out[(size_t)dst * 16 + l15], acc[r]);
    }
}

// ---------------------------------------------------------------------------
extern "C" void kernel_launch(void* const* d_in, const int* in_sizes, int n_in,
                              void* d_out, int out_size, void* d_ws, size_t ws_size,
                              hipStream_t stream) {
    const float* x      = (const float*)d_in[0];
    const int*   eidx   = (const int*)  d_in[1];
    const float* eattr  = (const float*)d_in[2];
    const float* gamma  = (const float*)d_in[3];
    const float* beta   = (const float*)d_in[4];
    const float* w_edge = (const float*)d_in[5];
    const float* b_edge = (const float*)d_in[6];
    const float* root   = (const float*)d_in[7];
    const float* bias   = (const float*)d_in[8];
    float* out = (float*)d_out;

    const int N = in_sizes[0] / 16;
    const int E = in_sizes[2] / 8;

    float* h_ws = (float*)d_ws;                    // N*16 floats
    float* wbig = h_ws + (size_t)N * 16;           // 144*16 floats

    build_wbig<<<1, 256, 0, stream>>>(w_edge, b_edge, wbig);
    node_pre<<<(N + 255) / 256, 256, 0, stream>>>(x, gamma, beta, root, bias,
                                                  h_ws, out, N);
    const int tiles  = (E + TILE - 1) / TILE;
    const int blocks = (tiles + EDGE_WAVES - 1) / EDGE_WAVES;
    edge_conv<<<blocks, 256, 0, stream>>>(eidx, eattr, h_ws, wbig, out, E);
}